// GATv2_19799799234951
// MI455X (gfx1250) — compile-verified
//
#include <hip/hip_runtime.h>
#include <hip/hip_bf16.h>
#include <math.h>

#define NODES      50000
#define EDGES      800000
#define INDIM      16
#define HID        64
#define NH         4
#define NGRAPH     128
#define ODIM       32
#define NLAYERS    8
#define FEATS      (NH * HID)      /* 256 */
#define NEG_SLOPE  0.2f

typedef float v2f __attribute__((ext_vector_type(2)));
typedef float v8f __attribute__((ext_vector_type(8)));

__device__ __forceinline__ v8f wmma_f32(v2f a, v2f b, v8f c) {
  // D = A(16x4 f32) * B(4x16 f32) + C(16x16 f32)
  return __builtin_amdgcn_wmma_f32_16x16x4_f32(
      /*neg_a=*/false, a, /*neg_b=*/false, b,
      /*c_mod=*/(short)0, c, /*reuse_a=*/false, /*reuse_b=*/false);
}

// ---------------------------------------------------------------- h0 GEMM
// h0 = feat[N,16] @ W_in[16,64] + b_in ; also h = h0.
// grid.x = N/16 row tiles, block = 128 (4 waves, one 16-col tile each).
__global__ void k_gemm_h0(const float* __restrict__ feat,
                          const float* __restrict__ W_in,
                          const float* __restrict__ b_in,
                          float* __restrict__ h0,
                          float* __restrict__ h) {
  const int lane  = threadIdx.x & 31;
  const int wv    = threadIdx.x >> 5;           // 0..3 -> column tile
  const int kHalf = lane >> 4;                  // 0/1
  const int mA    = blockIdx.x * 16 + (lane & 15);
  const int nCol  = wv * 16 + (lane & 15);
  v8f acc = {0.f, 0.f, 0.f, 0.f, 0.f, 0.f, 0.f, 0.f};
#pragma unroll
  for (int k0 = 0; k0 < INDIM; k0 += 4) {
    const int ka = k0 + 2 * kHalf;
    v2f a, b;
    a.x = feat[mA * INDIM + ka];
    a.y = feat[mA * INDIM + ka + 1];
    b.x = W_in[(ka    ) * HID + nCol];
    b.y = W_in[(ka + 1) * HID + nCol];
    acc = wmma_f32(a, b, acc);
  }
  const int   mBase = blockIdx.x * 16 + 8 * kHalf;
  const float bias  = b_in[nCol];
#pragma unroll
  for (int r = 0; r < 8; ++r) {
    const float v = acc[r] + bias;
    h0[(mBase + r) * HID + nCol] = v;
    h [(mBase + r) * HID + nCol] = v;
  }
}

// ---------------------------------------------------------------- layer GEMM
// fs = [h|h0] @ W_src + b_src ; fd = [h|h0] @ W_dst + b_dst
// grid = (N/16, 2), block = 256 (8 waves). Wave handles one 16-col tile of
// BOTH fs and fd (A fragment reused). K = 128.
__global__ void k_proj(const float* __restrict__ h,
                       const float* __restrict__ h0,
                       const float* __restrict__ W_src,
                       const float* __restrict__ b_src,
                       const float* __restrict__ W_dst,
                       const float* __restrict__ b_dst,
                       float* __restrict__ fs,
                       float* __restrict__ fd) {
  const int lane  = threadIdx.x & 31;
  const int wv    = threadIdx.x >> 5;                 // 0..7
  const int nTile = blockIdx.y * 8 + wv;              // 0..15
  const int nCol  = nTile * 16 + (lane & 15);         // 0..255
  const int mA    = blockIdx.x * 16 + (lane & 15);
  const int kHalf = lane >> 4;
  v8f accS = {0.f, 0.f, 0.f, 0.f, 0.f, 0.f, 0.f, 0.f};
  v8f accD = {0.f, 0.f, 0.f, 0.f, 0.f, 0.f, 0.f, 0.f};
#pragma unroll 4
  for (int k0 = 0; k0 < 2 * HID; k0 += 4) {
    const int ka = k0 + 2 * kHalf;                    // ka,ka+1 same half
    v2f a, bs, bd;
    if (ka < HID) {
      a.x = h[mA * HID + ka];
      a.y = h[mA * HID + ka + 1];
    } else {
      a.x = h0[mA * HID + (ka - HID)];
      a.y = h0[mA * HID + (ka - HID) + 1];
    }
    bs.x = W_src[(ka    ) * FEATS + nCol];
    bs.y = W_src[(ka + 1) * FEATS + nCol];
    bd.x = W_dst[(ka    ) * FEATS + nCol];
    bd.y = W_dst[(ka + 1) * FEATS + nCol];
    accS = wmma_f32(a, bs, accS);
    accD = wmma_f32(a, bd, accD);
  }
  const int   mBase = blockIdx.x * 16 + 8 * kHalf;
  const float bS = b_src[nCol], bD = b_dst[nCol];
#pragma unroll
  for (int r = 0; r < 8; ++r) {
    fs[(mBase + r) * FEATS + nCol] = accS[r] + bS;
    fd[(mBase + r) * FEATS + nCol] = accD[r] + bD;
  }
}

// ---------------------------------------------------------------- edge logits
// s[e,h] = sum_d leakyrelu(fs[src[e],h,d] + fd[dst[e],h,d]) * attn[h,d]
__global__ void k_edge_s(const float* __restrict__ fs,
                         const float* __restrict__ fd,
                         const int* __restrict__ src,
                         const int* __restrict__ dst,
                         const float* __restrict__ attn,
                         float* __restrict__ s) {
  const int tid = blockIdx.x * blockDim.x + threadIdx.x;
  if (tid >= EDGES * NH) return;
  const int e  = tid >> 2;
  const int hh = tid & 3;
  const float4* ps = (const float4*)(fs + (size_t)src[e] * FEATS + hh * HID);
  const float4* pd = (const float4*)(fd + (size_t)dst[e] * FEATS + hh * HID);
  const float4* pa = (const float4*)(attn + hh * HID);
  float acc = 0.f;
#pragma unroll
  for (int i = 0; i < HID / 4; ++i) {
    const float4 u = ps[i], v = pd[i], w = pa[i];
    float t;
    t = u.x + v.x; t = t > 0.f ? t : NEG_SLOPE * t; acc = fmaf(t, w.x, acc);
    t = u.y + v.y; t = t > 0.f ? t : NEG_SLOPE * t; acc = fmaf(t, w.y, acc);
    t = u.z + v.z; t = t > 0.f ? t : NEG_SLOPE * t; acc = fmaf(t, w.z, acc);
    t = u.w + v.w; t = t > 0.f ? t : NEG_SLOPE * t; acc = fmaf(t, w.w, acc);
  }
  s[tid] = acc;
}

// ---------------------------------------------------------------- per-node
// One wave per node. Lanes split the 64 dims (2 each). For each head:
// exact max over incoming edges, exp-sum, weighted agg; then tanh + head sum.
// Writes h in place (h only consumed by k_proj which already ran).
__global__ void k_node(const float* __restrict__ fs,
                       const float* __restrict__ s,
                       const int* __restrict__ src,
                       const int* __restrict__ row_ptr,
                       const int* __restrict__ edge_idx,
                       float* __restrict__ h) {
  const int lane = threadIdx.x & 31;
  const int node = blockIdx.x * (blockDim.x >> 5) + (threadIdx.x >> 5);
  if (node >= NODES) return;
  const int beg = row_ptr[node];
  const int end = row_ptr[node + 1];
  const int d0  = lane * 2;
  float o0 = 0.f, o1 = 0.f;
  if (end > beg) {
    for (int hh = 0; hh < NH; ++hh) {
      float m = -3.0e38f;
      for (int i = beg; i < end; ++i)
        m = fmaxf(m, s[edge_idx[i] * NH + hh]);
      float sum = 0.f, a0 = 0.f, a1 = 0.f;
      for (int i = beg; i < end; ++i) {
        const int   e = edge_idx[i];
        const float a = expf(s[e * NH + hh] - m);
        sum += a;
        const float2* p2 =
            (const float2*)(fs + (size_t)src[e] * FEATS + hh * HID);
        const float2 v = p2[lane];
        a0 = fmaf(a, v.x, a0);
        a1 = fmaf(a, v.y, a1);
      }
      o0 += tanhf(a0 / sum);
      o1 += tanhf(a1 / sum);
    }
  }
  h[node * HID + d0]     = o0;
  h[node * HID + d0 + 1] = o1;
}

// ---------------------------------------------------------------- CSR build
__global__ void k_zero_i(int* __restrict__ p, int n) {
  const int t = blockIdx.x * blockDim.x + threadIdx.x;
  if (t < n) p[t] = 0;
}
__global__ void k_zero_f(float* __restrict__ p, int n) {
  const int t = blockIdx.x * blockDim.x + threadIdx.x;
  if (t < n) p[t] = 0.f;
}
__global__ void k_count(const int* __restrict__ dst, int* __restrict__ deg) {
  const int e = blockIdx.x * blockDim.x + threadIdx.x;
  if (e < EDGES) atomicAdd(&deg[dst[e]], 1);
}
__global__ void k_scan(const int* __restrict__ deg,
                       int* __restrict__ row_ptr,
                       int* __restrict__ cursor) {
  __shared__ int sm[1024];
  const int t  = threadIdx.x;
  const int CH = (NODES + 1023) / 1024;   // 49
  int sum = 0;
  for (int i = 0; i < CH; ++i) {
    const int idx = t * CH + i;
    if (idx < NODES) sum += deg[idx];
  }
  sm[t] = sum;
  __syncthreads();
  for (int off = 1; off < 1024; off <<= 1) {
    int v = 0;
    if (t >= off) v = sm[t - off];
    __syncthreads();
    sm[t] += v;
    __syncthreads();
  }
  int run = (t == 0) ? 0 : sm[t - 1];
  for (int i = 0; i < CH; ++i) {
    const int idx = t * CH + i;
    if (idx < NODES) {
      row_ptr[idx] = run;
      cursor[idx]  = run;
      run += deg[idx];
    }
  }
  if (t == 0) row_ptr[NODES] = EDGES;
}
__global__ void k_fill(const int* __restrict__ dst,
                       int* __restrict__ cursor,
                       int* __restrict__ edge_idx) {
  const int e = blockIdx.x * blockDim.x + threadIdx.x;
  if (e < EDGES) {
    const int p = atomicAdd(&cursor[dst[e]], 1);
    edge_idx[p] = e;
  }
}

// ---------------------------------------------------------------- readout
__global__ void k_readout(const float* __restrict__ h,
                          const int* __restrict__ gids,
                          const int* __restrict__ isroot,
                          float* __restrict__ hg) {
  const int tid = blockIdx.x * blockDim.x + threadIdx.x;
  if (tid >= NODES * HID) return;
  const int n = tid / HID;
  const int d = tid - n * HID;
  if (isroot[n]) atomicAdd(&hg[gids[n] * HID + d], h[tid]);
}
__global__ void k_out(const float* __restrict__ hg,
                      const float* __restrict__ W_out,
                      const float* __restrict__ b_out,
                      float* __restrict__ out) {
  const int tid = blockIdx.x * blockDim.x + threadIdx.x;
  if (tid >= NGRAPH * ODIM) return;
  const int g = tid / ODIM;
  const int o = tid - g * ODIM;
  float acc = b_out[o];
#pragma unroll 8
  for (int k = 0; k < HID; ++k)
    acc = fmaf(hg[g * HID + k], W_out[k * ODIM + o], acc);
  out[tid] = acc;
}

// ---------------------------------------------------------------- launch
extern "C" void kernel_launch(void* const* d_in, const int* in_sizes, int n_in,
                              void* d_out, int out_size, void* d_ws, size_t ws_size,
                              hipStream_t stream) {
  const float* feat  = (const float*)d_in[0];
  const int*   src   = (const int*)  d_in[1];
  const int*   dst   = (const int*)  d_in[2];
  const int*   gids  = (const int*)  d_in[3];
  const int*   isrt  = (const int*)  d_in[4];
  const float* W_in  = (const float*)d_in[5];
  const float* b_in  = (const float*)d_in[6];
  const float* W_src = (const float*)d_in[7];
  const float* b_src = (const float*)d_in[8];
  const float* W_dst = (const float*)d_in[9];
  const float* b_dst = (const float*)d_in[10];
  const float* attn  = (const float*)d_in[11];
  const float* W_out = (const float*)d_in[12];
  const float* b_out = (const float*)d_in[13];
  float* out = (float*)d_out;

  char* w = (char*)d_ws;
  auto alloc = [&](size_t bytes) -> char* {
    char* p = w;
    w += (bytes + 255) & ~(size_t)255;
    return p;
  };
  float* h0       = (float*)alloc((size_t)NODES * HID * sizeof(float));
  float* h        = (float*)alloc((size_t)NODES * HID * sizeof(float));
  float* fs       = (float*)alloc((size_t)NODES * FEATS * sizeof(float));
  float* fd       = (float*)alloc((size_t)NODES * FEATS * sizeof(float));
  float* s        = (float*)alloc((size_t)EDGES * NH * sizeof(float));
  int*   deg      = (int*)  alloc((size_t)NODES * sizeof(int));
  int*   row_ptr  = (int*)  alloc((size_t)(NODES + 1) * sizeof(int));
  int*   cursor   = (int*)  alloc((size_t)NODES * sizeof(int));
  int*   edge_idx = (int*)  alloc((size_t)EDGES * sizeof(int));
  float* hg       = (float*)alloc((size_t)NGRAPH * HID * sizeof(float));

  const int rowTiles = NODES / 16;                 // 3125 (exact)

  // CSR by destination (topology fixed across layers)
  k_zero_i<<<(NODES + 255) / 256, 256, 0, stream>>>(deg, NODES);
  k_count <<<(EDGES + 255) / 256, 256, 0, stream>>>(dst, deg);
  k_scan  <<<1, 1024, 0, stream>>>(deg, row_ptr, cursor);
  k_fill  <<<(EDGES + 255) / 256, 256, 0, stream>>>(dst, cursor, edge_idx);

  // h0 = feat @ W_in + b_in ; h = h0
  k_gemm_h0<<<rowTiles, 128, 0, stream>>>(feat, W_in, b_in, h0, h);

  for (int layer = 0; layer < NLAYERS; ++layer) {
    k_proj<<<dim3(rowTiles, 2), 256, 0, stream>>>(h, h0, W_src, b_src,
                                                  W_dst, b_dst, fs, fd);
    k_edge_s<<<(EDGES * NH + 255) / 256, 256, 0, stream>>>(fs, fd, src, dst,
                                                           attn, s);
    k_node<<<(NODES + 7) / 8, 256, 0, stream>>>(fs, s, src, row_ptr,
                                                edge_idx, h);
  }

  k_zero_f<<<(NGRAPH * HID + 255) / 256, 256, 0, stream>>>(hg, NGRAPH * HID);
  k_readout<<<(NODES * HID + 255) / 256, 256, 0, stream>>>(h, gids, isrt, hg);
  k_out<<<(NGRAPH * ODIM + 255) / 256, 256, 0, stream>>>(hg, W_out, b_out, out);
}